// GraphConvolution_81965155877088
// MI455X (gfx1250) — compile-verified
//
#include <hip/hip_runtime.h>

typedef __attribute__((ext_vector_type(2))) float v2f;
typedef __attribute__((ext_vector_type(8))) float v8f;

#define WAVES_PER_BLOCK 8

// ---------------------------------------------------------------------------
// H[n_rows x 64] = X[n_rows x K] * W[K x 64]   (W row-major [K][64])
// One wave (32 lanes) computes a 16-row x 64-col tile with V_WMMA_F32_16X16X4_F32.
// A layout (16x4 f32): lanes 0-15 = rows, VGPR0/1 = K {0,1}; lanes 16-31 = K {2,3}.
// B layout (4x16 f32): symmetric; C/D: VGPR i -> M=i (lanes 0-15), M=8+i (lanes 16-31).
// W is staged in LDS transposed with +4 padding so both A and B read as float2.
// n_rows is assumed to be a multiple of 16 (100000 = 6250*16 here).
// ---------------------------------------------------------------------------
__global__ __launch_bounds__(256) void gemm_wmma_f32(
    const float* __restrict__ X, const float* __restrict__ W,
    float* __restrict__ H, int n_rows, int K) {
  extern __shared__ float sWt[];  // [64][K+4], transposed, padded
  const int t = threadIdx.x;
  const int stride = K + 4;
  for (int i = t; i < K * 64; i += 256) {
    const int k = i >> 6;   // row of W
    const int n = i & 63;   // col of W
    sWt[n * stride + k] = W[i];
  }
  __syncthreads();

  const int wave = t >> 5;
  const int lane = t & 31;
  const long row0 = ((long)blockIdx.x * WAVES_PER_BLOCK + wave) * 16;
  if (row0 >= n_rows) return;

  const int m  = lane & 15;         // row within tile (A) / col within n-tile (B,D)
  const int kk = (lane >> 4) * 2;   // lane-group K offset: 0 or 2

  const float* xrow = X + (row0 + m) * (long)K;
  v8f acc0 = {}, acc1 = {}, acc2 = {}, acc3 = {};

  for (int k = 0; k < K; k += 4) {
    const v2f a  = *(const v2f*)(xrow + k + kk);
    const v2f b0 = *(const v2f*)(sWt + ( 0 + m) * stride + k + kk);
    const v2f b1 = *(const v2f*)(sWt + (16 + m) * stride + k + kk);
    const v2f b2 = *(const v2f*)(sWt + (32 + m) * stride + k + kk);
    const v2f b3 = *(const v2f*)(sWt + (48 + m) * stride + k + kk);
    acc0 = __builtin_amdgcn_wmma_f32_16x16x4_f32(false, a, false, b0, (short)0, acc0, false, false);
    acc1 = __builtin_amdgcn_wmma_f32_16x16x4_f32(false, a, false, b1, (short)0, acc1, false, false);
    acc2 = __builtin_amdgcn_wmma_f32_16x16x4_f32(false, a, false, b2, (short)0, acc2, false, false);
    acc3 = __builtin_amdgcn_wmma_f32_16x16x4_f32(false, a, false, b3, (short)0, acc3, false, false);
  }

  // D layout: VGPR i: lanes 0-15 -> M=i, N=lane; lanes 16-31 -> M=8+i, N=lane-16
  const int moff = (lane >> 4) * 8;
  float* hrow = H + (row0 + moff) * 64 + m;
#pragma unroll
  for (int i = 0; i < 8; ++i) {
    hrow[(long)i * 64 +  0] = acc0[i];
    hrow[(long)i * 64 + 16] = acc1[i];
    hrow[(long)i * 64 + 32] = acc2[i];
    hrow[(long)i * 64 + 48] = acc3[i];
  }
}

// ---------------------------------------------------------------------------
// One wave per edge: agg[dst] += w * H[src]  (64 floats/row, 2 per lane).
// h / agg are each 25.6 MB -> L2-resident on MI455X (192 MB L2), so gathers and
// float atomics are served by L2, not HBM. unsafeAtomicAdd forces the native
// global_atomic_add_f32 (no CAS loop) — this phase is the roofline bottleneck.
// ---------------------------------------------------------------------------
__global__ __launch_bounds__(256) void edge_scatter(
    const float* __restrict__ H, const int* __restrict__ src,
    const int* __restrict__ dst, const float* __restrict__ ew,
    float* __restrict__ agg, int n_edges) {
  const int lane = threadIdx.x & 31;
  const long e = ((long)blockIdx.x * blockDim.x + threadIdx.x) >> 5;
  if (e >= n_edges) return;
  const int   s = src[e];
  const int   d = dst[e];
  const float w = ew[e];
  const float* hs = H + (long)s * 64;
  float* ad = agg + (long)d * 64;
  __builtin_prefetch(ad + lane);            // gfx1250 global_prefetch_b8
  const float v0 = hs[lane]      * w;
  const float v1 = hs[lane + 32] * w;
  unsafeAtomicAdd(ad + lane,      v0);
  unsafeAtomicAdd(ad + lane + 32, v1);
}

__global__ __launch_bounds__(256) void zero_f4(float4* __restrict__ p, long n4) {
  const long i = (long)blockIdx.x * blockDim.x + threadIdx.x;
  if (i < n4) p[i] = make_float4(0.f, 0.f, 0.f, 0.f);
}

// out = relu(in + bias[col]), 4 floats per thread, row width 64
__global__ __launch_bounds__(256) void bias_relu(
    const float* __restrict__ in, const float* __restrict__ bias,
    float* __restrict__ out, long n4) {
  const long i = (long)blockIdx.x * blockDim.x + threadIdx.x;
  if (i >= n4) return;
  float4 v = ((const float4*)in)[i];
  const float4 b = *(const float4*)(bias + (int)((i * 4) & 63));
  v.x = fmaxf(v.x + b.x, 0.f);
  v.y = fmaxf(v.y + b.y, 0.f);
  v.z = fmaxf(v.z + b.z, 0.f);
  v.w = fmaxf(v.w + b.w, 0.f);
  ((float4*)out)[i] = v;
}

extern "C" void kernel_launch(void* const* d_in, const int* in_sizes, int n_in,
                              void* d_out, int out_size, void* d_ws, size_t ws_size,
                              hipStream_t stream) {
  const float* x    = (const float*)d_in[0];   // [n_nodes,128]
  const int*   ei   = (const int*)  d_in[1];   // [2, n_edges]
  const float* mask = (const float*)d_in[2];   // [n_edges]
  const float* W1   = (const float*)d_in[3];   // [128,64]
  const float* b1   = (const float*)d_in[4];   // [64]
  const float* W2   = (const float*)d_in[5];   // [64,64]
  const float* b2   = (const float*)d_in[6];   // [64]

  const int n_nodes = in_sizes[0] / 128;
  const int n_edges = in_sizes[2];
  const int* src = ei;
  const int* dst = ei + n_edges;

  float* h   = (float*)d_ws;                   // [n_nodes,64]
  float* agg = h + (size_t)n_nodes * 64;       // [n_nodes,64]
  float* out = (float*)d_out;                  // [n_nodes,64]

  const long n64 = (long)n_nodes * 64;
  const long n4  = n64 / 4;

  const dim3 blk(256);
  const int tiles       = (n_nodes + 15) / 16;
  const int gemm_blocks = (tiles + WAVES_PER_BLOCK - 1) / WAVES_PER_BLOCK;
  const int edge_blocks = (int)(((long)n_edges * 32 + 255) / 256);
  const int elem_blocks = (int)((n4 + 255) / 256);

  // ---------------- layer 1 ----------------
  gemm_wmma_f32<<<gemm_blocks, blk, 64 * (128 + 4) * sizeof(float), stream>>>(
      x, W1, h, n_nodes, 128);
  zero_f4<<<elem_blocks, blk, 0, stream>>>((float4*)agg, n4);
  edge_scatter<<<edge_blocks, blk, 0, stream>>>(h, src, dst, mask, agg, n_edges);
  bias_relu<<<elem_blocks, blk, 0, stream>>>(agg, b1, h, n4);   // h = relu(agg+b1)

  // ---------------- layer 2 ----------------
  gemm_wmma_f32<<<gemm_blocks, blk, 64 * (64 + 4) * sizeof(float), stream>>>(
      h, W2, agg, n_nodes, 64);                                 // agg = h @ W2
  zero_f4<<<elem_blocks, blk, 0, stream>>>((float4*)out, n4);
  edge_scatter<<<edge_blocks, blk, 0, stream>>>(agg, src, dst, mask, out, n_edges);
  bias_relu<<<elem_blocks, blk, 0, stream>>>(out, b2, out, n4); // out = relu(out+b2)
}